// UnifiedKAN_54889682043456
// MI455X (gfx1250) — compile-verified
//
#include <hip/hip_runtime.h>

// ---------------------------------------------------------------------------
// UnifiedKAN on MI455X (gfx1250, wave32).
//
//  * kan_prep <<<20, 32>>>: block b converts spline b (16 L1 edges + 4 L2)
//    into 6 piecewise cubics; one Cox-de Boor eval per thread (6 intervals x
//    4 Newton nodes), combined via LDS. Block 0's spare threads compute the
//    softmax weights + base offset.
//  * kan_main <<<n/1024, 256>>>: one TDM tensor_load_to_lds stages the
//    parameter block into LDS per workgroup (CDNA5 async-tensor path).
//    Grid-uniform values (interval boundaries, softmax weights, base offset)
//    are read via uniform global loads -> scalar s_load (K$), so the only
//    per-lane LDS traffic is the j-dependent interval-param fetch
//    (ds_load_b128 + ds_load_b64, 32B-aligned). Each thread streams 4
//    samples. Branchless range clamp matches the reference's S(x)=0 for
//    x<-1 or x>=1 (dropped 9th basis function).
//    Global traffic 10.5 MB -> ~0.45 us at 23.3 TB/s; ~300 VALU/sample.
//
// ws/LDS layout (floats):
//   spline s (0..19) at s*64:
//     [0..4]  interval boundaries i0..i4   [5..7] pad
//     [8 + j*8 .. ] per interval j (0..5): {lo, 1/h, a0, a1, a2, a3, 0, 0}
//   [1280..1287] softmax w[h][0..1]   [1288] base_offset   [..1295] pad
// ---------------------------------------------------------------------------

#define KAN_DEG   3
#define KAN_NIVL  6
#define KAN_IN    4
#define KAN_HID   4
#define KAN_NSPL  20
#define SPL_STRIDE 64
#define WS_W      1280
#define WS_BASE   1288
#define WS_TOTAL  1296
#define SPB       4          // samples per thread in kan_main

typedef __attribute__((ext_vector_type(4))) unsigned int u32x4_;
typedef __attribute__((ext_vector_type(8))) int          i32x8_;
typedef __attribute__((ext_vector_type(4))) int          i32x4_;

// --- Cox-de Boor oracle: mirrors the reference b_spline_basis + dot --------
__device__ __forceinline__ float kan_cdb(const float t[13],
                                         const float* __restrict__ cf, float x)
{
    float b[12];
#pragma unroll
    for (int k = 0; k < 12; ++k)
        b[k] = (x >= t[k] && x < t[k + 1]) ? 1.0f : 0.0f;
    if (x == t[12]) b[11] += 1.0f;
#pragma unroll
    for (int d = 1; d <= KAN_DEG; ++d) {
#pragma unroll
        for (int k = 0; k < 12 - d; ++k) {
            float den1 = t[k + d] - t[k];
            float den2 = t[k + d + 1] - t[k + 1];
            float t1 = (den1 > 1e-8f) ? (x - t[k]) / den1 * b[k] : 0.0f;
            float t2 = (den2 > 1e-8f) ? (t[k + d + 1] - x) / den2 * b[k + 1] : 0.0f;
            b[k] = t1 + t2;
        }
    }
    float v = 0.0f;
#pragma unroll
    for (int k = 0; k < 8; ++k) v += b[k] * cf[k];   // 9th basis dropped (c8=0)
    return v;
}

__device__ __forceinline__ float kan_lo_of(const float itr[5], int j)
{
    float lo = -1.0f;
    lo = (j >= 1) ? itr[0] : lo;  lo = (j >= 2) ? itr[1] : lo;
    lo = (j >= 3) ? itr[2] : lo;  lo = (j >= 4) ? itr[3] : lo;
    lo = (j >= 5) ? itr[4] : lo;
    return lo;
}
__device__ __forceinline__ float kan_hi_of(const float itr[5], int j)
{
    float hi = 1.0f;
    hi = (j <= 4) ? itr[4] : hi;  hi = (j <= 3) ? itr[3] : hi;
    hi = (j <= 2) ? itr[2] : hi;  hi = (j <= 1) ? itr[1] : hi;
    hi = (j <= 0) ? itr[0] : hi;
    return hi;
}

__global__ void __launch_bounds__(32)
kan_prep(const float* __restrict__ l1_c, const float* __restrict__ l1_kd,
         const float* __restrict__ alpha, const float* __restrict__ gum,
         const float* __restrict__ l2_c, const float* __restrict__ l2_kd,
         const float* __restrict__ base, float* __restrict__ ws)
{
    const int spl = blockIdx.x;          // one spline per block (0..19)
    const int tid = threadIdx.x;
    __shared__ float Y[KAN_NIVL * 4];

    const float* kd = (spl < 16) ? (l1_kd + spl * 6) : (l2_kd + (spl - 16) * 6);
    const float* cf = (spl < 16) ? (l1_c + spl * 8) : (l2_c + (spl - 16) * 8);

    // knot vector (computed redundantly by every thread)
    float d[6]; float sum = 0.0f;
#pragma unroll
    for (int m = 0; m < 6; ++m) {
        float v = kd[m];
        d[m] = fmaxf(v, 0.0f) + log1pf(expf(-fabsf(v)));   // stable softplus
        sum += d[m];
    }
    const float scale = 2.0f / sum;
    float itr[5]; float cum = -1.0f;
#pragma unroll
    for (int m = 0; m < 5; ++m) { cum += d[m] * scale; itr[m] = cum; }
    float t[13];
    t[0] = t[1] = t[2] = t[3] = -1.0f;
    t[4] = itr[0]; t[5] = itr[1]; t[6] = itr[2]; t[7] = itr[3]; t[8] = itr[4];
    t[9] = t[10] = t[11] = t[12] = 1.0f;

    // one Cox-de Boor evaluation per thread: interval j, Newton node m
    if (tid < KAN_NIVL * 4) {
        const int j = tid >> 2, m = tid & 3;
        float lo = kan_lo_of(itr, j), hi = kan_hi_of(itr, j);
        float h  = (hi - lo) * 0.25f;
        Y[tid] = kan_cdb(t, cf, lo + (float)m * h);
    }
    // boundaries + pads (independent of Y)
    if (tid < 5)        ws[spl * SPL_STRIDE + tid] = itr[tid];
    else if (tid < 8)   ws[spl * SPL_STRIDE + tid] = 0.0f;
    __syncthreads();

    // combine 4 node values -> power-basis cubic (exact for a cubic)
    if (tid < KAN_NIVL) {
        const int j = tid;
        float y0 = Y[4 * j], y1 = Y[4 * j + 1], y2 = Y[4 * j + 2], y3 = Y[4 * j + 3];
        float lo = kan_lo_of(itr, j), hi = kan_hi_of(itr, j);
        float h  = (hi - lo) * 0.25f;
        float d1 = y1 - y0;
        float d2 = y2 - 2.0f * y1 + y0;
        float d3 = y3 - 3.0f * y2 + 3.0f * y1 - y0;
        float* q = ws + spl * SPL_STRIDE + 8 + j * 8;
        q[0] = lo;
        q[1] = 1.0f / h;                               // x -> s in [0,4)
        q[2] = y0;                                     // a0
        q[3] = d1 - 0.5f * d2 + (1.0f / 3.0f) * d3;    // a1
        q[4] = 0.5f * (d2 - d3);                       // a2
        q[5] = d3 * (1.0f / 6.0f);                     // a3
        q[6] = 0.0f; q[7] = 0.0f;
    }

    if (spl == 0) {
        if (tid >= 24 && tid < 24 + KAN_HID) {         // softmax, TAU = 1
            const int hh = tid - 24;
            float e0 = alpha[2 * hh] + gum[2 * hh];
            float e1 = alpha[2 * hh + 1] + gum[2 * hh + 1];
            float mx = fmaxf(e0, e1);
            float w0 = expf(e0 - mx), w1 = expf(e1 - mx);
            float inv = 1.0f / (w0 + w1);
            ws[WS_W + 2 * hh]     = w0 * inv;
            ws[WS_W + 2 * hh + 1] = w1 * inv;
        }
        if (tid == 28) {
            ws[WS_BASE] = base[0];
#pragma unroll
            for (int k = WS_BASE + 1; k < WS_TOTAL; ++k) ws[k] = 0.0f;
        }
    }
}

// --- streaming evaluation: boundaries arrive in SGPRs (uniform loads) -------
__device__ __forceinline__ float kan_eval_x(const float4 bb, float b5,
                                            const float* __restrict__ S, float x)
{
    int j = (x >= bb.x) + (x >= bb.y) + (x >= bb.z) + (x >= bb.w) + (x >= b5);
    const float* c = S + 8 + j * 8;                 // 32B-aligned LDS
    float4 c0 = *(const float4*)c;                  // lo, 1/h, a0, a1
    float2 c1 = *(const float2*)(c + 4);            // a2, a3
    float s = (x - c0.x) * c0.y;
    float v = fmaf(fmaf(fmaf(c1.y, s, c1.x), s, c0.w), s, c0.z);
    return (x < -1.0f || x >= 1.0f) ? 0.0f : v;     // reference range semantics
}

__global__ void __launch_bounds__(256)
kan_main(const float4* __restrict__ x, const float* __restrict__ ws,
         float* __restrict__ out, int n)
{
    __shared__ __align__(16) float P[WS_TOTAL];

#if defined(__gfx1250__) && __has_builtin(__builtin_amdgcn_tensor_load_to_lds)
    if (threadIdx.x == 0) {
        // TDM descriptor: 1-row tensor, WS_TOTAL x f32, global ws -> LDS P.
        unsigned lds = (unsigned)(size_t)(&P[0]);   // flat LDS addr low 32 bits
        unsigned long long ga = (unsigned long long)(size_t)ws;
        u32x4_ g0 = {
            1u,                                               // count=1, valid
            lds,                                              // lds_addr
            (unsigned)(ga & 0xFFFFFFFFu),                     // global_addr lo
            (unsigned)((ga >> 32) & 0x1FFFFFFu) | (2u << 30)  // ga[56:32]|type=2
        };
        i32x8_ g1 = {
            (int)(2u << 16),                          // wg_mask=0, data_size=4B
            (int)((WS_TOTAL & 0xFFFF) << 16),         // tensor_dim0 lo16
            (int)((WS_TOTAL >> 16) | (1 << 16)),      // dim0 hi16 | tensor_dim1=1
            (int)(((unsigned)WS_TOTAL) << 16),        // dim1 hi16=0 | tile_dim0
            1,                                        // tile_dim1=1, tile_dim2=0
            WS_TOTAL,                                 // tensor_dim0_stride lo32
            0, 0
        };
        i32x4_ gz4 = { 0, 0, 0, 0 };
        i32x8_ gz8 = { 0, 0, 0, 0, 0, 0, 0, 0 };
        __builtin_amdgcn_tensor_load_to_lds(g0, g1, gz4, gz4, gz8, 0);
        __builtin_amdgcn_s_wait_tensorcnt(0);
    }
    __syncthreads();
#else
    for (int i = threadIdx.x; i < WS_TOTAL; i += blockDim.x) P[i] = ws[i];
    __syncthreads();
#endif

    const int nt = gridDim.x * blockDim.x;
    const int t0 = blockIdx.x * blockDim.x + threadIdx.x;

    int   idx[SPB];
    bool  ok[SPB];
    float xin[SPB][KAN_IN];
#pragma unroll
    for (int u = 0; u < SPB; ++u) {
        idx[u] = t0 + u * nt;
        ok[u]  = idx[u] < n;
        float4 xv = ok[u] ? x[idx[u]] : make_float4(0.f, 0.f, 0.f, 0.f);
        xin[u][0] = xv.x; xin[u][1] = xv.y; xin[u][2] = xv.z; xin[u][3] = xv.w;
    }

    float acc[SPB];
    const float bo = ws[WS_BASE];                  // uniform -> s_load
#pragma unroll
    for (int u = 0; u < SPB; ++u) acc[u] = bo;

#pragma unroll
    for (int h = 0; h < KAN_HID; ++h) {
        float sumv[SPB], prodv[SPB];
#pragma unroll
        for (int u = 0; u < SPB; ++u) { sumv[u] = 0.0f; prodv[u] = 1.0f; }
#pragma unroll
        for (int ii = 0; ii < KAN_IN; ++ii) {
            const int spl = h * KAN_IN + ii;
            const float* S = &P[spl * SPL_STRIDE];                 // LDS params
            float4 bb = *(const float4*)(ws + spl * SPL_STRIDE);   // uniform -> s_load_b128
            float  b5 = ws[spl * SPL_STRIDE + 4];                  // uniform -> s_load_b32
#pragma unroll
            for (int u = 0; u < SPB; ++u) {
                float v = kan_eval_x(bb, b5, S, xin[u][ii]);
                sumv[u]  += v;
                prodv[u] *= fminf(fmaxf(v, -5.0f), 5.0f);
            }
        }
        const float w0 = ws[WS_W + 2 * h], w1 = ws[WS_W + 2 * h + 1];  // uniform
        const int spl2 = 16 + h;
        const float* S2 = &P[spl2 * SPL_STRIDE];
        float4 bb2 = *(const float4*)(ws + spl2 * SPL_STRIDE);
        float  b52 = ws[spl2 * SPL_STRIDE + 4];
#pragma unroll
        for (int u = 0; u < SPB; ++u) {
            float hid = w0 * sumv[u] + w1 * prodv[u];
            acc[u] += kan_eval_x(bb2, b52, S2, hid);
        }
    }
#pragma unroll
    for (int u = 0; u < SPB; ++u)
        if (ok[u]) out[idx[u]] = acc[u];
}

extern "C" void kernel_launch(void* const* d_in, const int* in_sizes, int n_in,
                              void* d_out, int out_size, void* d_ws, size_t ws_size,
                              hipStream_t stream)
{
    (void)n_in; (void)out_size; (void)ws_size;
    const float* x     = (const float*)d_in[0];
    const float* l1_c  = (const float*)d_in[1];
    const float* l1_kd = (const float*)d_in[2];
    const float* alpha = (const float*)d_in[3];
    const float* gum   = (const float*)d_in[4];
    const float* l2_c  = (const float*)d_in[5];
    const float* l2_kd = (const float*)d_in[6];
    const float* base  = (const float*)d_in[7];
    float* ws  = (float*)d_ws;
    float* out = (float*)d_out;

    const int n = in_sizes[0] / KAN_IN;                  // N = 524288
    const int blocks = (n + 256 * SPB - 1) / (256 * SPB);

    kan_prep<<<KAN_NSPL, 32, 0, stream>>>(l1_c, l1_kd, alpha, gum, l2_c, l2_kd,
                                          base, ws);
    kan_main<<<blocks, 256, 0, stream>>>((const float4*)x, ws, out, n);
}